// TPNet_22539988369627
// MI455X (gfx1250) — compile-verified
//
#include <hip/hip_runtime.h>

#define NODES   65536
#define EDGES   1048576
#define BATCH   16
#define KDER    5
#define TSTEPS  20
#define CHUNK   2048            // edges staged per block iteration (16 KB LDS)

typedef __attribute__((ext_vector_type(2))) float v2f;
typedef __attribute__((ext_vector_type(8))) float v8f;
typedef __attribute__((ext_vector_type(2))) int   v2i;

#if defined(__gfx1250__) && __has_builtin(__builtin_amdgcn_global_load_async_to_lds_b64)
#define HAVE_ASYNC_LDS 1
typedef __attribute__((address_space(1))) v2i* gptr_v2i;   // global
typedef __attribute__((address_space(3))) v2i* lptr_v2i;   // LDS
#else
#define HAVE_ASYNC_LDS 0
#endif

// ---------------------------------------------------------------------------
// w[e] = sum_k dist[e,k]*coeffs[k]  via V_WMMA_F32_16X16X4_F32.
// One wave handles 16 edges: A = dist[e0:e0+16, 0:4] (16x4), B has every
// column equal to coeffs[0:4] (4x16) so every column of D replicates the
// 16 dot products; the K=5 remainder is added at writeout.
// ---------------------------------------------------------------------------
__global__ void compute_w_wmma(const float* __restrict__ dist,
                               const float* __restrict__ coeffs,
                               float* __restrict__ w) {
    const int tid  = threadIdx.x;
    const int wave = (blockIdx.x << 3) + (tid >> 5);   // 8 waves per block
    const int lane = tid & 31;
    const int half = lane >> 4;                        // 0: K=0,1   1: K=2,3
    const int m    = lane & 15;
    const int e0   = wave << 4;                        // 16 edges per wave

    const float* de = dist + (size_t)(e0 + m) * KDER + (half << 1);
    v2f a; a[0] = de[0];                a[1] = de[1];
    v2f b; b[0] = coeffs[half << 1];    b[1] = coeffs[(half << 1) + 1];
    v8f c = {0.f, 0.f, 0.f, 0.f, 0.f, 0.f, 0.f, 0.f};

    v8f d = __builtin_amdgcn_wmma_f32_16x16x4_f32(
        false, a, false, b, (short)0, c, false, false);

    if (m == 0) {                 // lane 0 -> edges e0..e0+7, lane 16 -> e0+8..e0+15
        const float c4   = coeffs[4];
        const int   base = e0 + (half << 3);
        #pragma unroll
        for (int v = 0; v < 8; ++v)
            w[base + v] = d[v] + dist[(size_t)(base + v) * KDER + 4] * c4;
    }
}

// ---------------------------------------------------------------------------
// CSR construction: histogram -> exclusive scan -> scatter
// ---------------------------------------------------------------------------
__global__ void zero_ints(int* p) {
    p[blockIdx.x * 256 + threadIdx.x] = 0;
}

__global__ void hist_kernel(const int* __restrict__ dst, int* __restrict__ counts) {
    const int e = blockIdx.x * 256 + threadIdx.x;
    atomicAdd(&counts[dst[e]], 1);
}

__global__ void scan_local(const int* __restrict__ in, int* __restrict__ out,
                           int* __restrict__ bsums) {
    __shared__ int s[256];
    const int t = threadIdx.x;
    const int g = blockIdx.x * 256 + t;
    const int v = in[g];
    s[t] = v;
    __syncthreads();
    for (int d = 1; d < 256; d <<= 1) {
        const int add = (t >= d) ? s[t - d] : 0;
        __syncthreads();
        s[t] += add;
        __syncthreads();
    }
    out[g] = s[t] - v;                               // exclusive scan
    if (bsums && t == 255) bsums[blockIdx.x] = s[255];
}

__global__ void scan_add(int* __restrict__ offs, const int* __restrict__ boffs,
                         int* __restrict__ cursor) {
    const int g = blockIdx.x * 256 + threadIdx.x;
    const int v = offs[g] + boffs[blockIdx.x];
    offs[g]   = v;
    cursor[g] = v;
}

__global__ void scatter_kernel(const int* __restrict__ src, const int* __restrict__ dst,
                               const float* __restrict__ w, int* __restrict__ cursor,
                               int2* __restrict__ csr) {
    const int e   = blockIdx.x * 256 + threadIdx.x;
    const int pos = atomicAdd(&cursor[dst[e]], 1);
    csr[pos] = make_int2(src[e], __float_as_int(w[e]));
}

// ---------------------------------------------------------------------------
// x[B,NODES] -> node-major x[NODES][16] (one-time, 4 MB)
// ---------------------------------------------------------------------------
__global__ void transpose_in(const float* __restrict__ xin, float* __restrict__ xA) {
    const int idx = blockIdx.x * 256 + threadIdx.x;
    const int b = idx & 15, n = idx >> 4;
    xA[idx] = xin[(size_t)b * NODES + n];
}

// ---------------------------------------------------------------------------
// One Euler step. Block = 64 nodes x 4 channel-quads. A block's 64 nodes own
// one contiguous CSR segment; stage it into LDS (async global->LDS on CDNA5)
// once, then all 4 quads consume it from LDS -> CSR global traffic /4.
// After the scatter kernel, cursor[n] == row end, offs[n] == row start.
// ---------------------------------------------------------------------------
__global__ void step_kernel(const float* __restrict__ xcur, float* __restrict__ xnext,
                            const int* __restrict__ offs, const int* __restrict__ rowend,
                            const int2* __restrict__ csr, float* __restrict__ out,
                            const float* __restrict__ dtp, int t) {
    __shared__ int2 sEdges[CHUNK];

    const int tid   = threadIdx.x;
    const int q     = tid >> 6;                      // channel quad 0..3
    const int node0 = blockIdx.x * 64;
    const int node  = node0 + (tid & 63);

    const int segStart = offs[node0];
    const int segEnd   = rowend[node0 + 63];
    const int start    = offs[node];
    const int end      = rowend[node];

    float4 acc = make_float4(0.f, 0.f, 0.f, 0.f);

    for (int c0 = segStart; c0 < segEnd; c0 += CHUNK) {
        const int n = min(CHUNK, segEnd - c0);

        // ---- cooperative stage csr[c0 .. c0+n) -> sEdges[0..n) ----
#if HAVE_ASYNC_LDS
        for (int j = tid; j < n; j += 256) {
            __builtin_amdgcn_global_load_async_to_lds_b64(
                (gptr_v2i)(csr + c0 + j),
                (lptr_v2i)(sEdges + j),
                0, 0);
        }
        __builtin_amdgcn_s_wait_asynccnt(0);
#else
        for (int j = tid; j < n; j += 256)
            sEdges[j] = csr[c0 + j];
#endif
        __syncthreads();

        // ---- consume my row's slice of this chunk ----
        const int lo = max(start - c0, 0);
        const int hi = min(end - c0, n);
        for (int i = lo; i < hi; ++i) {
            const int2  ew = sEdges[i];
            const float wv = __int_as_float(ew.y);
            const float4 xv =
                *reinterpret_cast<const float4*>(xcur + ((size_t)ew.x << 4) + (q << 2));
            acc.x = fmaf(wv, xv.x, acc.x);
            acc.y = fmaf(wv, xv.y, acc.y);
            acc.z = fmaf(wv, xv.z, acc.z);
            acc.w = fmaf(wv, xv.w, acc.w);
        }
        __syncthreads();                             // LDS reuse guard
    }

    const float dt = dtp[0];
    const float4 x0 =
        *reinterpret_cast<const float4*>(xcur + ((size_t)node << 4) + (q << 2));
    float4 y;
    y.x = fmaf(dt, acc.x, x0.x);
    y.y = fmaf(dt, acc.y, x0.y);
    y.z = fmaf(dt, acc.z, x0.z);
    y.w = fmaf(dt, acc.w, x0.w);

    *reinterpret_cast<float4*>(xnext + ((size_t)node << 4) + (q << 2)) = y;

    const int b0 = q << 2;                           // out layout [B, T, NODES]
    out[((size_t)(b0 + 0) * TSTEPS + t) * NODES + node] = y.x;
    out[((size_t)(b0 + 1) * TSTEPS + t) * NODES + node] = y.y;
    out[((size_t)(b0 + 2) * TSTEPS + t) * NODES + node] = y.z;
    out[((size_t)(b0 + 3) * TSTEPS + t) * NODES + node] = y.w;
}

// ---------------------------------------------------------------------------
extern "C" void kernel_launch(void* const* d_in, const int* in_sizes, int n_in,
                              void* d_out, int out_size, void* d_ws, size_t ws_size,
                              hipStream_t stream) {
    const float* x_in   = (const float*)d_in[0];
    const int*   eidx   = (const int*)  d_in[2];
    const float* dtp    = (const float*)d_in[3];
    const float* dist   = (const float*)d_in[5];
    const float* coeffs = (const float*)d_in[6];
    float*       out    = (float*)d_out;

    const int* srcp = eidx;
    const int* dstp = eidx + EDGES;

    char* ws = (char*)d_ws;                       // ~21 MB used
    float* w      = (float*)(ws);                                  // 4 MB
    int2*  csr    = (int2*) (ws + ((size_t)4  << 20));             // 8 MB
    int*   counts = (int*)  (ws + ((size_t)12 << 20));             // 256 KB
    int*   offs   = (int*)  (ws + ((size_t)12 << 20) + (256 << 10));
    int*   cursor = (int*)  (ws + ((size_t)12 << 20) + (512 << 10));
    int*   bsums  = (int*)  (ws + ((size_t)12 << 20) + (768 << 10));
    int*   boffs  = bsums + 256;
    float* xA     = (float*)(ws + ((size_t)13 << 20));             // 4 MB
    float* xB     = (float*)(ws + ((size_t)17 << 20));             // 4 MB

    // Per-edge weights (WMMA) and CSR build
    compute_w_wmma<<<EDGES / 128, 256, 0, stream>>>(dist, coeffs, w);
    zero_ints     <<<NODES / 256, 256, 0, stream>>>(counts);
    hist_kernel   <<<EDGES / 256, 256, 0, stream>>>(dstp, counts);
    scan_local    <<<NODES / 256, 256, 0, stream>>>(counts, offs, bsums);
    scan_local    <<<1,           256, 0, stream>>>(bsums, boffs, nullptr);
    scan_add      <<<NODES / 256, 256, 0, stream>>>(offs, boffs, cursor);
    scatter_kernel<<<EDGES / 256, 256, 0, stream>>>(srcp, dstp, w, cursor, csr);

    // Node-major state, then 20 sequential Euler steps (ping-pong)
    transpose_in<<<(NODES * BATCH) / 256, 256, 0, stream>>>(x_in, xA);
    float* cur = xA;
    float* nxt = xB;
    for (int t = 0; t < TSTEPS; ++t) {
        step_kernel<<<NODES / 64, 256, 0, stream>>>(cur, nxt, offs, cursor, csr,
                                                    out, dtp, t);
        float* tmp = cur; cur = nxt; nxt = tmp;
    }
}